// AttentiveAtlasEncoder_85804856640017
// MI455X (gfx1250) — compile-verified
//
#include <hip/hip_runtime.h>
#include <math.h>

// Problem constants
#define B_   4096
#define IN_  128
#define H_   256
#define D_   32
#define C_   8
#define K_   256
#define HS_  16

// d_out layout (floats), concatenated reference-return order
#define O_KCHART  0
#define O_KCODE   4096
#define O_ZN      8192
#define O_ZTEX    139264
#define O_ROUTER  270336
#define O_ZGEO    303104
#define O_VQ      434176
#define O_INDICES 434177
#define O_ZNALL   466945
#define O_CBAR    1515521
#define O_VLOCAL  1646593

typedef __attribute__((ext_vector_type(2))) float v2f;
typedef __attribute__((ext_vector_type(8))) float v8f;

__device__ __forceinline__ float gelu_exact(float x) {
    return 0.5f * x * (1.0f + erff(x * 0.70710678118654752f));
}

// ---------------------------------------------------------------------------
// Fused GEMM + bias + optional exact-GELU using V_WMMA_F32_16X16X4_F32.
// One wave computes one 16x16 output tile; K stepped by 4.
// ---------------------------------------------------------------------------
__global__ __launch_bounds__(256)
void aae_gemm_bias_act(const float* __restrict__ A, const float* __restrict__ Bm,
                       const float* __restrict__ bias, float* __restrict__ Out,
                       int M, int N, int K, int act) {
    const int waveId = (blockIdx.x * blockDim.x + threadIdx.x) >> 5;
    const int lane   = threadIdx.x & 31;
    const int tilesN = N >> 4;
    const int tm = (waveId / tilesN) << 4;
    const int tn = (waveId % tilesN) << 4;
    const int hl = lane >> 4;      // lane half (0 or 1)
    const int lm = lane & 15;

    v8f acc = {0.f, 0.f, 0.f, 0.f, 0.f, 0.f, 0.f, 0.f};
    const float* arow = A + (size_t)(tm + lm) * K;

    for (int k0 = 0; k0 < K; k0 += 4) {
        // A: lane L holds A[lm][k0 + 2*hl + r], r=0,1 -> contiguous pair
        v2f a = *(const v2f*)(arow + k0 + 2 * hl);
        // B: lane L holds B[k0 + 2*hl + r][tn+lm]
        const float* bp = Bm + (size_t)(k0 + 2 * hl) * N + tn + lm;
        v2f b; b.x = bp[0]; b.y = bp[N];
        acc = __builtin_amdgcn_wmma_f32_16x16x4_f32(
                false, a, false, b, (short)0, acc, false, false);
    }

    const float bb = bias[tn + lm];
#pragma unroll
    for (int r = 0; r < 8; ++r) {
        const int m = hl * 8 + r;            // D row held in VGPR r, lane-half hl
        float v = acc[r] + bb;
        if (act) v = gelu_exact(v);
        Out[(size_t)(tm + m) * N + tn + lm] = v;
    }
}

// ---------------------------------------------------------------------------
// Per-row router: scores -> softmax -> argmax, c_bar, v_local
// ---------------------------------------------------------------------------
__global__ __launch_bounds__(256)
void aae_router(const float* __restrict__ v, const float* __restrict__ centers,
                float* __restrict__ out) {
    const int b = blockIdx.x * blockDim.x + threadIdx.x;
    if (b >= B_) return;
    float vr[D_];
#pragma unroll
    for (int d = 0; d < D_; ++d) vr[d] = v[b * D_ + d];

    float sc[C_];
#pragma unroll
    for (int c = 0; c < C_; ++c) {
        float s = 0.f;
#pragma unroll
        for (int d = 0; d < D_; ++d) s = fmaf(vr[d], centers[c * D_ + d], s);
        sc[c] = s * 0.17677669529663687f;   // 1/sqrt(32)
    }
    float mx = sc[0]; int am = 0;
#pragma unroll
    for (int c = 1; c < C_; ++c) if (sc[c] > mx) { mx = sc[c]; am = c; }
    float e[C_], sum = 0.f;
#pragma unroll
    for (int c = 0; c < C_; ++c) { e[c] = expf(sc[c] - mx); sum += e[c]; }
    const float inv = 1.f / sum;
    float r[C_];
#pragma unroll
    for (int c = 0; c < C_; ++c) { r[c] = e[c] * inv; out[O_ROUTER + b * C_ + c] = r[c]; }
    out[O_KCHART + b] = (float)am;

    float cb[D_];
#pragma unroll
    for (int d = 0; d < D_; ++d) cb[d] = 0.f;
#pragma unroll
    for (int c = 0; c < C_; ++c)
#pragma unroll
        for (int d = 0; d < D_; ++d) cb[d] = fmaf(r[c], centers[c * D_ + d], cb[d]);
#pragma unroll
    for (int d = 0; d < D_; ++d) {
        out[O_CBAR + b * D_ + d]   = cb[d];
        out[O_VLOCAL + b * D_ + d] = vr[d] - cb[d];
    }
}

// ---------------------------------------------------------------------------
// VQ search: block = (32 rows, 1 chart). Chart codebook staged transposed in
// LDS (stride 257 -> conflict-free across the 64 banks). Wave w handles rows
// w*4..w*4+3; per row each lane scans 8 codes, then wave32 shfl_xor argmin.
// ---------------------------------------------------------------------------
#define VQ_ROWS 32
__global__ __launch_bounds__(256)
void aae_vq(const float* __restrict__ codebook, float* __restrict__ out,
            float* __restrict__ zq_all) {
    __shared__ float cbT[D_][K_ + 1];        // [d][k], 32x257
    __shared__ float vls[VQ_ROWS][D_ + 1];
    const int c = blockIdx.y;
    const int row0 = blockIdx.x * VQ_ROWS;
    const int tid = threadIdx.x;

    const float* cb = codebook + (size_t)c * K_ * D_;
    for (int q = tid; q < K_ * D_; q += 256) {
        int k = q >> 5, d = q & 31;
        cbT[d][k] = cb[q];
    }
    for (int q = tid; q < VQ_ROWS * D_; q += 256) {
        int r = q >> 5, d = q & 31;
        vls[r][d] = out[O_VLOCAL + (row0 + r) * D_ + d];
    }
    __syncthreads();

    const int wave = tid >> 5, lane = tid & 31;
#pragma unroll
    for (int rr = 0; rr < 4; ++rr) {
        const int r = wave * 4 + rr;
        float best = 3.4e38f; int bestk = 0;
        for (int j = 0; j < 8; ++j) {
            const int k = lane + j * 32;
            float s = 0.f;
#pragma unroll
            for (int d = 0; d < D_; ++d) {
                const float df = vls[r][d] - cbT[d][k];
                s = fmaf(df, df, s);
            }
            if (s < best) { best = s; bestk = k; }
        }
        // wave32 argmin, first-occurrence tie-break
#pragma unroll
        for (int off = 16; off >= 1; off >>= 1) {
            const float ob = __shfl_xor(best, off, 32);
            const int   ok = __shfl_xor(bestk, off, 32);
            if (ob < best || (ob == best && ok < bestk)) { best = ob; bestk = ok; }
        }
        const int row = row0 + r;
        if (lane == 0) out[O_INDICES + row * C_ + c] = (float)bestk;
        zq_all[(size_t)(row * C_ + c) * D_ + lane] = cbT[lane][bestk];  // lane == d
    }
}

// ---------------------------------------------------------------------------
// z_n_all = gelu(delta @ Ws1 + bs1) @ Ws2 + bs2, one thread per (b,c)
// ---------------------------------------------------------------------------
__global__ __launch_bounds__(256)
void aae_znall(const float* __restrict__ zq_all,
               const float* __restrict__ Ws1, const float* __restrict__ bs1,
               const float* __restrict__ Ws2, const float* __restrict__ bs2,
               float* __restrict__ out) {
    __shared__ float sW1[D_ * HS_], sW2[HS_ * D_], sB1[HS_], sB2[D_];
    const int tid = threadIdx.x;
    for (int i = tid; i < D_ * HS_; i += 256) { sW1[i] = Ws1[i]; sW2[i] = Ws2[i]; }
    if (tid < HS_) sB1[tid] = bs1[tid];
    if (tid < D_)  sB2[tid] = bs2[tid];
    __syncthreads();

    const int t = blockIdx.x * blockDim.x + tid;   // 0..B*C-1
    const int b = t >> 3;
    float delta[D_];
#pragma unroll
    for (int d = 0; d < D_; ++d)
        delta[d] = out[O_VLOCAL + b * D_ + d] - zq_all[(size_t)t * D_ + d];
    float h[HS_];
#pragma unroll
    for (int j = 0; j < HS_; ++j) {
        float s = sB1[j];
#pragma unroll
        for (int d = 0; d < D_; ++d) s = fmaf(delta[d], sW1[d * HS_ + j], s);
        h[j] = gelu_exact(s);
    }
#pragma unroll
    for (int d = 0; d < D_; ++d) {
        float s = sB2[d];
#pragma unroll
        for (int j = 0; j < HS_; ++j) s = fmaf(h[j], sW2[j * D_ + d], s);
        out[O_ZNALL + (size_t)t * D_ + d] = s;
    }
}

// ---------------------------------------------------------------------------
// Finalize per row: z_n, z_tex, z_geo, K_code, per-row vq-loss partial
// ---------------------------------------------------------------------------
__global__ __launch_bounds__(256)
void aae_finalize(float* __restrict__ out, const float* __restrict__ zq_all,
                  float* __restrict__ rowloss) {
    const int b = blockIdx.x * blockDim.x + threadIdx.x;
    if (b >= B_) return;
    float r[C_];
#pragma unroll
    for (int c = 0; c < C_; ++c) r[c] = out[O_ROUTER + b * C_ + c];
    float vl[D_], zn[D_], zqb[D_];
#pragma unroll
    for (int d = 0; d < D_; ++d) { vl[d] = out[O_VLOCAL + b * D_ + d]; zn[d] = 0.f; zqb[d] = 0.f; }
#pragma unroll
    for (int c = 0; c < C_; ++c) {
        const float* zr = zq_all + (size_t)(b * C_ + c) * D_;
        const float* nr = out + O_ZNALL + (size_t)(b * C_ + c) * D_;
#pragma unroll
        for (int d = 0; d < D_; ++d) {
            zqb[d] = fmaf(r[c], zr[d], zqb[d]);
            zn[d]  = fmaf(r[c], nr[d], zn[d]);
        }
    }
#pragma unroll
    for (int d = 0; d < D_; ++d) {
        const float cb = out[O_CBAR + b * D_ + d];
        out[O_ZN + b * D_ + d]   = zn[d];
        out[O_ZTEX + b * D_ + d] = vl[d] - zqb[d] - zn[d];
        out[O_ZGEO + b * D_ + d] = cb + zqb[d] + zn[d];     // c_bar + z_q_st + z_n (fwd)
    }
    const int kch = (int)out[O_KCHART + b];
    out[O_KCODE + b] = out[O_INDICES + b * C_ + kch];

    // vq loss partial: d_cb == d_cm in forward, loss = 1.25/B * sum d^2 * w
    float vn2 = 0.f;
#pragma unroll
    for (int d = 0; d < D_; ++d) vn2 = fmaf(vl[d], vl[d], vn2);
    const float nrm = fmaxf(sqrtf(vn2), 1e-6f);
    const float scale = fminf(0.99f / nrm, 1.0f);
    float yp[D_], yn = 0.f;
#pragma unroll
    for (int d = 0; d < D_; ++d) { yp[d] = vl[d] * scale; yn = fmaf(yp[d], yp[d], yn); }
    float acc = 0.f;
#pragma unroll
    for (int c = 0; c < C_; ++c) {
        const float* zr = zq_all + (size_t)(b * C_ + c) * D_;
        float sq = 0.f, xn = 0.f;
#pragma unroll
        for (int d = 0; d < D_; ++d) {
            const float df = zr[d] - yp[d];
            sq = fmaf(df, df, sq);
            xn = fmaf(zr[d], zr[d], xn);
        }
        const float denom = fmaxf((1.f - xn) * (1.f - yn), 1e-6f);
        const float arg = fmaxf(1.f + 2.f * sq / denom, 1.f + 1e-6f);
        const float dd = acoshf(arg);
        acc = fmaf(dd * dd, r[c], acc);
    }
    rowloss[b] = acc;
}

// Deterministic fixed-order reduction of 4096 partials -> vq_loss scalar
__global__ __launch_bounds__(256)
void aae_reduce_loss(const float* __restrict__ rowloss, float* __restrict__ out) {
    __shared__ float s[256];
    const int t = threadIdx.x;
    float a = 0.f;
    for (int i = t; i < B_; i += 256) a += rowloss[i];
    s[t] = a; __syncthreads();
    for (int off = 128; off >= 1; off >>= 1) {
        if (t < off) s[t] += s[t + off];
        __syncthreads();
    }
    if (t == 0) out[O_VQ] = s[0] * (1.25f / (float)B_);
}

// ---------------------------------------------------------------------------
extern "C" void kernel_launch(void* const* d_in, const int* in_sizes, int n_in,
                              void* d_out, int out_size, void* d_ws, size_t ws_size,
                              hipStream_t stream) {
    const float* x        = (const float*)d_in[0];
    const float* W1       = (const float*)d_in[1];
    const float* b1       = (const float*)d_in[2];
    const float* W2       = (const float*)d_in[3];
    const float* b2       = (const float*)d_in[4];
    const float* Wv       = (const float*)d_in[5];
    const float* bv       = (const float*)d_in[6];
    const float* centers  = (const float*)d_in[7];
    const float* codebook = (const float*)d_in[8];
    const float* Ws1      = (const float*)d_in[9];
    const float* bs1      = (const float*)d_in[10];
    const float* Ws2      = (const float*)d_in[11];
    const float* bs2      = (const float*)d_in[12];
    float* out = (float*)d_out;

    float* ws = (float*)d_ws;
    float* feats1  = ws;                       // 4096*256
    float* feats2  = ws + 1048576;             // 4096*256
    float* v       = ws + 2097152;             // 4096*32
    float* zq_all  = ws + 2228224;             // 4096*8*32
    float* rowloss = ws + 3276800;             // 4096

    // feats1 = gelu(x @ W1 + b1): 4096x128x256 -> 4096 WMMA tiles
    aae_gemm_bias_act<<<512, 256, 0, stream>>>(x, W1, b1, feats1, B_, H_, IN_, 1);
    // feats2 = gelu(feats1 @ W2 + b2): 4096x256x256
    aae_gemm_bias_act<<<512, 256, 0, stream>>>(feats1, W2, b2, feats2, B_, H_, H_, 1);
    // v = feats2 @ Wv + bv: 4096x256x32
    aae_gemm_bias_act<<<64, 256, 0, stream>>>(feats2, Wv, bv, v, B_, D_, H_, 0);
    // router / K_chart / c_bar / v_local
    aae_router<<<16, 256, 0, stream>>>(v, centers, out);
    // VQ nearest-code search (LDS-staged codebook)
    aae_vq<<<dim3(B_ / VQ_ROWS, C_), 256, 0, stream>>>(codebook, out, zq_all);
    // refinement MLP per (b, chart)
    aae_znall<<<(B_ * C_) / 256, 256, 0, stream>>>(zq_all, Ws1, bs1, Ws2, bs2, out);
    // blended outputs + per-row loss partials
    aae_finalize<<<16, 256, 0, stream>>>(out, zq_all, rowloss);
    // deterministic scalar reduction
    aae_reduce_loss<<<1, 256, 0, stream>>>(rowloss, out);
}